// KECRS_66881230733493
// MI455X (gfx1250) — compile-verified
//
#include <hip/hip_runtime.h>
#include <hip/hip_bf16.h>
#include <math.h>

#define N_ENTITY 64368
#define N_REL    12
#define DIM      128
#define NB       8
#define N_ITEM   6924
#define BATCH    128
#define SEQL     32

typedef __attribute__((ext_vector_type(2))) float v2f;
typedef __attribute__((ext_vector_type(8))) float v8f;

// ---------------------------------------------------------------------------
// Kernel 1: per-edge RGCN message scatter.
// One wave32 per edge; lane owns 4 consecutive d-values (float4 = 16B load).
// msg[d] = sum_b att[rt,b] * basis[b, src, d]   (avoids materializing 396MB w)
// aggr[dst, d] += msg[d]  via f32 atomics (aggr is 33MB -> L2 resident)
// ---------------------------------------------------------------------------
__global__ void rgcn_edge_scatter(const int* __restrict__ edge_idx,
                                  const int* __restrict__ edge_type,
                                  const float* __restrict__ basis,
                                  const float* __restrict__ att,
                                  float* __restrict__ aggr,
                                  float* __restrict__ deg,
                                  int nedges) {
  const int wid = blockIdx.x * (blockDim.x >> 5) + (threadIdx.x >> 5);
  if (wid >= nedges) return;                       // wave-uniform exit
  const int lane = threadIdx.x & 31;
  const int src = edge_idx[wid];
  const int dst = edge_idx[nedges + wid];
  const int rt  = edge_type[wid];

  float coeff[NB];
#pragma unroll
  for (int b = 0; b < NB; ++b) coeff[b] = att[rt * NB + b];

  const int d0 = lane * 4;
  float4 acc = make_float4(0.f, 0.f, 0.f, 0.f);
#pragma unroll
  for (int b = 0; b < NB; ++b) {
    const float4 bv =
        *(const float4*)(basis + ((size_t)b * N_ENTITY + src) * DIM + d0);
    acc.x = fmaf(coeff[b], bv.x, acc.x);
    acc.y = fmaf(coeff[b], bv.y, acc.y);
    acc.z = fmaf(coeff[b], bv.z, acc.z);
    acc.w = fmaf(coeff[b], bv.w, acc.w);
  }
  float* out = aggr + (size_t)dst * DIM + d0;
  atomicAdd(out + 0, acc.x);
  atomicAdd(out + 1, acc.y);
  atomicAdd(out + 2, acc.z);
  atomicAdd(out + 3, acc.w);
  if (lane == 0) atomicAdd(deg + dst, 1.0f);
}

// ---------------------------------------------------------------------------
// Kernel 2: nodes = aggr / max(deg,1) + root + bias   (in place over aggr)
// ---------------------------------------------------------------------------
__global__ void rgcn_finalize(float* __restrict__ aggr,
                              const float* __restrict__ deg,
                              const float* __restrict__ root,
                              const float* __restrict__ bias) {
  const int idx = blockIdx.x * blockDim.x + threadIdx.x;
  if (idx >= N_ENTITY * DIM) return;
  const int n = idx >> 7;
  const int d = idx & (DIM - 1);
  aggr[idx] = aggr[idx] / fmaxf(deg[n], 1.0f) + root[idx] + bias[d];
}

// ---------------------------------------------------------------------------
// Kernel 3: masked attention pooling. One block (256 thr = 8 waves) per batch
// row. h tile staged in LDS (rows padded to DIM+1 floats: row stride 129 is
// coprime with the 64 LDS banks, so the A-fragment column reads for WMMA are
// conflict-free). tanh(h @ attn_a) runs on the matrix pipe as f32 WMMA
// 16x16x4 with A fed from LDS; epilogue reduces tanh(t)*attn_b across each
// 16-lane half via wave32 shuffles -> one LDS atomic per output row half.
// ---------------------------------------------------------------------------
__global__ void attn_pool(const float* __restrict__ nodes,
                          const int* __restrict__ seed_ids,
                          const int* __restrict__ seed_len,
                          const float* __restrict__ attn_a,
                          const float* __restrict__ attn_b,
                          float* __restrict__ u) {
  __shared__ float h[SEQL][DIM + 1];   // +1 pad: kill 16-way bank conflicts
  __shared__ float ev[SEQL];
  __shared__ float aw[SEQL];
  __shared__ int   sid[SEQL];

  const int b   = blockIdx.x;
  const int tid = threadIdx.x;
  const int len = seed_len[b];

  if (tid < SEQL) {
    sid[tid] = seed_ids[b * SEQL + tid];
    ev[tid] = 0.f;
  }
  __syncthreads();

  for (int p = tid; p < SEQL * DIM; p += blockDim.x) {
    const int l = p >> 7, d = p & (DIM - 1);
    h[l][d] = nodes[(size_t)sid[l] * DIM + d];
  }
  __syncthreads();

  // t = h @ attn_a  (32x128 = 2 x 8 tiles of 16x16), then
  // e[m] += sum_n tanh(t[m,n]) * attn_b[n]
  const int wave = tid >> 5;
  const int lane = tid & 31;
  const int half = lane >> 4;
  const int l15  = lane & 15;
  for (int t = wave; t < 16; t += 8) {             // 2 tiles per wave, uniform
    const int m0 = (t >> 3) * 16;
    const int n0 = (t & 7) * 16;
    v8f c = {};
#pragma unroll
    for (int k = 0; k < DIM; k += 4) {
      const int kk = k + 2 * half;
      v2f a, bm;
      a.x  = h[m0 + l15][kk];                      // ds_load (LDS-fed WMMA)
      a.y  = h[m0 + l15][kk + 1];
      bm.x = attn_a[kk * DIM + n0 + l15];          // B[k,n] = attn_a[k,n]
      bm.y = attn_a[(kk + 1) * DIM + n0 + l15];
      c = __builtin_amdgcn_wmma_f32_16x16x4_f32(false, a, false, bm, (short)0,
                                                c, false, false);
    }
    const float wb = attn_b[n0 + l15];
#pragma unroll
    for (int r = 0; r < 8; ++r) {
      float contrib = tanhf(c[r]) * wb;
#pragma unroll
      for (int o = 8; o; o >>= 1)                  // reduce within 16-lane half
        contrib += __shfl_xor(contrib, o, 32);
      if (l15 == 0) atomicAdd(&ev[m0 + r + 8 * half], contrib);
    }
  }
  __syncthreads();

  // softmax over l with mask (l < len), exactly mirroring where(mask, e, -1e9)
  if (tid < 32) {
    float val = (tid < len) ? ev[tid] : -1e9f;
    float m = val;
#pragma unroll
    for (int o = 16; o; o >>= 1) m = fmaxf(m, __shfl_xor(m, o, 32));
    const float ex = expf(val - m);
    float s = ex;
#pragma unroll
    for (int o = 16; o; o >>= 1) s += __shfl_xor(s, o, 32);
    aw[tid] = ex / s;
  }
  __syncthreads();

  if (tid < DIM) {
    float acc = 0.f;
#pragma unroll
    for (int l = 0; l < SEQL; ++l) acc = fmaf(aw[l], h[l][tid], acc);
    u[b * DIM + tid] = (len > 0) ? acc : 0.f;
  }
}

// ---------------------------------------------------------------------------
// Kernel 4: scores = u @ nodes[:N_ITEM]^T + out_bias, via f32 WMMA 16x16x4.
// 128 threads = 4 waves per block; each wave owns one 16x16 output tile and
// iterates K=128 in steps of 4 (32 unrolled v_wmma_f32_16x16x4_f32).
// A lane layout (ISA 7.12.2): lanes 0-15 M=lane hold K={k,k+1}; lanes 16-31
// hold K={k+2,k+3}. B mirrored over N; C/D: 8 VGPRs, M split by lane half.
// Partial last N-tile: clamp load index (EXEC stays all-1s), guard stores.
// ---------------------------------------------------------------------------
__global__ void scores_wmma(const float* __restrict__ u,
                            const float* __restrict__ nodes,
                            const float* __restrict__ out_bias,
                            float* __restrict__ out) {
  const int lane  = threadIdx.x & 31;
  const int wave  = threadIdx.x >> 5;
  const int ntile = blockIdx.x * 4 + wave;
  const int n0    = ntile * 16;
  if (n0 >= N_ITEM) return;                        // wave-uniform exit
  const int m0   = blockIdx.y * 16;
  const int half = lane >> 4;
  const int l15  = lane & 15;
  const int mA   = m0 + l15;
  const int nB   = n0 + l15;
  const int nC   = (nB < N_ITEM) ? nB : (N_ITEM - 1);

  v8f c = {};
#pragma unroll
  for (int k = 0; k < DIM; k += 4) {
    const int kk = k + 2 * half;
    v2f a, bm;
    a.x  = u[mA * DIM + kk];
    a.y  = u[mA * DIM + kk + 1];
    bm.x = nodes[(size_t)nC * DIM + kk];
    bm.y = nodes[(size_t)nC * DIM + kk + 1];
    c = __builtin_amdgcn_wmma_f32_16x16x4_f32(false, a, false, bm, (short)0, c,
                                              false, false);
  }

  if (nB < N_ITEM) {
    const float bias = out_bias[nB];
#pragma unroll
    for (int r = 0; r < 8; ++r) {
      const int m = m0 + r + 8 * half;
      out[(size_t)m * N_ITEM + nB] = c[r] + bias;
    }
  }
}

// ---------------------------------------------------------------------------
extern "C" void kernel_launch(void* const* d_in, const int* in_sizes, int n_in,
                              void* d_out, int out_size, void* d_ws,
                              size_t ws_size, hipStream_t stream) {
  const int*   edge_idx  = (const int*)d_in[0];
  const int*   edge_type = (const int*)d_in[1];
  const int*   seed_ids  = (const int*)d_in[2];
  const int*   seed_len  = (const int*)d_in[3];
  // d_in[4] = labels (unused by reference output)
  const float* basis     = (const float*)d_in[5];
  const float* att       = (const float*)d_in[6];
  const float* root      = (const float*)d_in[7];
  const float* rgcn_bias = (const float*)d_in[8];
  const float* attn_a    = (const float*)d_in[9];
  const float* attn_b    = (const float*)d_in[10];
  const float* out_bias  = (const float*)d_in[11];
  const int    nedges    = in_sizes[1];

  // workspace layout (~34 MB): [aggr/nodes N*D f32][deg N f32][u B*D f32]
  char*  ws        = (char*)d_ws;
  const size_t aggrBytes = (size_t)N_ENTITY * DIM * sizeof(float);
  const size_t degBytes  = (size_t)N_ENTITY * sizeof(float);
  float* aggr = (float*)ws;                        // becomes `nodes` in place
  float* deg  = (float*)(ws + aggrBytes);
  float* uBuf = (float*)(ws + aggrBytes + ((degBytes + 255) & ~(size_t)255));

  // zero atomic accumulators every call (graph replays re-run this memset)
  hipMemsetAsync(ws, 0, aggrBytes + degBytes, stream);

  {  // 1M edges, one wave32 per edge, 8 waves per block
    const int threads = 256, wpb = threads / 32;
    const int blocks = (nedges + wpb - 1) / wpb;
    rgcn_edge_scatter<<<blocks, threads, 0, stream>>>(
        edge_idx, edge_type, basis, att, aggr, deg, nedges);
  }
  {
    const int total = N_ENTITY * DIM;
    rgcn_finalize<<<(total + 255) / 256, 256, 0, stream>>>(aggr, deg, root,
                                                           rgcn_bias);
  }
  attn_pool<<<BATCH, 256, 0, stream>>>(aggr, seed_ids, seed_len, attn_a,
                                       attn_b, uBuf);
  {
    const int ntiles = (N_ITEM + 15) / 16;               // 433
    dim3 grid((ntiles + 3) / 4, BATCH / 16);             // (109, 8)
    scores_wmma<<<grid, 128, 0, stream>>>(uBuf, aggr, out_bias, (float*)d_out);
  }
}